// LogSlaterDet_44573170598546
// MI455X (gfx1250) — compile-verified
//
#include <hip/hip_runtime.h>
#include <cmath>

#define NS   4096   // batch
#define NP   128    // particles
#define NH   64     // per-spin block size
#define SD   3      // spatial dim (K, padded to 4 for WMMA)
#define STR  65     // padded LDS row stride (conflict-free column access)

typedef __attribute__((ext_vector_type(2))) float v2f;
typedef __attribute__((ext_vector_type(8))) float v8f;

__global__ __launch_bounds__(256)
void LogSlaterDet_44573170598546_kernel(const float* __restrict__ x,
                                        const float* __restrict__ Wup,
                                        const float* __restrict__ Wdn,
                                        float* __restrict__ out,
                                        float log_norm) {
    __shared__ float A[2][NH][STR];   // two 64x64 spin blocks, padded rows
    __shared__ float s_inv[2];
    __shared__ int   s_piv[2];
    __shared__ float s_ld[2];

    const int tid  = threadIdx.x;
    const int lane = tid & 31;
    const int wave = tid >> 5;        // 8 wave32s
    const int b    = blockIdx.x;

    // ---------------- Orbital GEMM via V_WMMA_F32_16X16X4_F32 ----------------
    // wave -> (spin, row-tile). Each wave: 1 A-fragment reused over 4 N-tiles.
    {
        const int spin = wave >> 2;          // 0: up, 1: dn
        const int mt   = wave & 3;           // 16-row tile within spin block
        const float* Xb = x + (size_t)b * (NP * SD) + (size_t)spin * (NH * SD);
        const float* W  = spin ? Wdn : Wup;  // (3, 64), orbital index fastest

        // K=3 zero-pad: clamp the address (always valid), mask the VALUE.
        // Unconditional loads -> no exec-mask branches around global_load.
        const int   k0    = (lane >> 4) * 2;                 // 0 or 2
        const int   ky    = (k0 + 1 < SD) ? (k0 + 1) : (SD - 1);
        const float ymask = (k0 + 1 < SD) ? 1.0f : 0.0f;

        // A fragment 16x4 f32: lanes 0-15 -> K={0,1}, lanes 16-31 -> K={2,3}
        const int arow = mt * 16 + (lane & 15);
        v2f a;
        a.x = Xb[arow * SD + k0];
        a.y = Xb[arow * SD + ky] * ymask;

        // Preload all four B fragments (4x16 f32 each) up front.
        const int ncol = lane & 15;
        v2f bv[4];
        #pragma unroll
        for (int nt = 0; nt < 4; ++nt) {
            bv[nt].x = W[k0 * NH + nt * 16 + ncol];
            bv[nt].y = W[ky * NH + nt * 16 + ncol] * ymask;
        }

        const int rbase = mt * 16 + 8 * (lane >> 4);         // D-tile layout
        #pragma unroll
        for (int nt = 0; nt < 4; ++nt) {
            v8f c = {};
            // (neg_a, A, neg_b, B, c_mod, C, reuse_a, reuse_b)
            c = __builtin_amdgcn_wmma_f32_16x16x4_f32(
                    false, a, false, bv[nt], (short)0, c, false, false);
            #pragma unroll
            for (int r = 0; r < 8; ++r)
                A[spin][rbase + r][nt * 16 + ncol] = c[r];
        }
    }
    __syncthreads();

    // -------- Two concurrent 64x64 LU factorizations (partial pivoting) -----
    // threads [0,128) -> A[0], threads [128,256) -> A[1]; identical barrier
    // counts in both halves, so workgroup barriers stay aligned.
    const int half = tid >> 7;
    const int t    = tid & 127;
    float (*M)[STR] = A[half];
    float acc = 0.0f;                 // log|det| accumulator (leader thread)

    for (int k = 0; k < NH; ++k) {
        // Pivot argmax over column k: leader wave of each half (t in [0,32)),
        // 2 rows per lane + 5-step wave32 butterfly reduction.
        if (t < 32) {
            float v0 = (t      >= k) ? fabsf(M[t][k])      : -1.0f;
            float v1 = (t + 32 >= k) ? fabsf(M[t + 32][k]) : -1.0f;
            float bvv = v0; int bi = t;
            if (v1 > bvv) { bvv = v1; bi = t + 32; }
            #pragma unroll
            for (int m = 16; m >= 1; m >>= 1) {
                float ov = __shfl_xor(bvv, m, 32);
                int   oi = __shfl_xor(bi,  m, 32);
                if (ov > bvv) { bvv = ov; bi = oi; }
            }
            if (t == 0) {
                const float piv = M[bi][k];
                acc += logf(fabsf(piv));
                s_piv[half] = bi;
                s_inv[half] = 1.0f / piv;
            }
        }
        __syncthreads();
        const int p = s_piv[half];
        if (p != k && t < NH) {       // row swap, one column element per thread
            float tmp = M[k][t]; M[k][t] = M[p][t]; M[p][t] = tmp;
        }
        __syncthreads();
        const int r  = t & 63;        // one row per thread,
        const int ch = t >> 6;        // two threads split the columns of a row
        if (r > k) {
            const float L = M[r][k] * s_inv[half];
            for (int j = k + 1 + ch; j < NH; j += 2)
                M[r][j] -= L * M[k][j];
        }
        __syncthreads();
    }

    if (t == 0) s_ld[half] = acc;
    __syncthreads();
    if (tid == 0) out[b] = s_ld[0] + s_ld[1] - log_norm;
}

extern "C" void kernel_launch(void* const* d_in, const int* in_sizes, int n_in,
                              void* d_out, int out_size, void* d_ws, size_t ws_size,
                              hipStream_t stream) {
    const float* x   = (const float*)d_in[0];   // (4096, 128, 3)
    const float* Wup = (const float*)d_in[1];   // (3, 64)
    const float* Wdn = (const float*)d_in[2];   // (3, 64)
    float* out = (float*)d_out;                 // (4096,)

    const float log_norm = (float)(0.5 * lgamma((double)(NP + 1)));
    const int grid = (out_size > 0) ? out_size : NS;  // one workgroup per sample

    LogSlaterDet_44573170598546_kernel<<<grid, 256, 0, stream>>>(
        x, Wup, Wdn, out, log_norm);
}